// ANN_20126216749990
// MI455X (gfx1250) — compile-verified
//
#include <hip/hip_runtime.h>
#include <hip/hip_bf16.h>

// Problem constants (from the reference)
#define BATCH 8192
#define FDIM  4200
#define TDIM  512
#define HDIM  10

#define ROWS_PER_BLOCK 16
#define NWAVES 8
#define BLOCK_THREADS (NWAVES * 32)
#define OUTSTR 516   // padded LDS row stride for outputs[16][512] (516 % 64 != 0 -> no bank conflicts)

#define NSTEP1 (FDIM / 4)   // 1050 k-steps for x @ W1
#define NSTEP3 (TDIM / 4)   // 128 k-steps for outputs @ W3

typedef __attribute__((ext_vector_type(2))) float v2f;
typedef __attribute__((ext_vector_type(8))) float v8f;

#define WMMA_F32_4(A, B, C) \
    __builtin_amdgcn_wmma_f32_16x16x4_f32(false, (A), false, (B), (short)0, (C), false, false)

__device__ __forceinline__ float leaky01(float v) {
    return v > 0.0f ? v : 0.01f * v;
}

// ---------------------------------------------------------------------------
// Prep kernel: swizzle W1 [F,10] and W3 [T,10] into per-step WMMA B-fragment
// layout (B 4x16 f32, zero-padded N=10..15):
//   frag[s*64 + lane*2 + c] = B[k = 4s + 2*(lane>=16) + c][n = lane%16]
// so the hot loop's B load is one coalesced global_load_b64 per lane.
// ---------------------------------------------------------------------------
__global__ __launch_bounds__(256)
void ANN_prep_frags(const float* __restrict__ W1, const float* __restrict__ W3,
                    float* __restrict__ w1f, float* __restrict__ w3f) {
    const int i = blockIdx.x * 256 + threadIdx.x;
    const int r = i & 63;
    const int lane = r >> 1;
    const int c = r & 1;
    const int n = lane & 15;
    const int khalf = (lane >> 4) ? 2 : 0;
    if (i < NSTEP1 * 64) {
        const int s = i >> 6;
        const int k = s * 4 + khalf + c;
        w1f[i] = (n < HDIM) ? W1[k * HDIM + n] : 0.0f;
    }
    if (i < NSTEP3 * 64) {
        const int s = i >> 6;
        const int k = s * 4 + khalf + c;
        w3f[i] = (n < HDIM) ? W3[k * HDIM + n] : 0.0f;
    }
}

// ---------------------------------------------------------------------------
// Fused kernel: 16 batch rows per workgroup, 8 waves.
//   Phase 1: h = leaky(x @ W1 + b1)      -> WMMA f32 16x16x4, K = 4200
//   Phase 2: q = sigmoid(h @ W2 + b2); spline gather + Horner -> LDS outputs
//   Phase 3: h2 = leaky(outputs @ W3 + b3); out = h2 @ W4 + b4 -> WMMA, K = 512
// ---------------------------------------------------------------------------
template <bool USE_FRAG>
__global__ __launch_bounds__(BLOCK_THREADS)
void ANN_spline_fused_kernel(const float* __restrict__ x,
                             const float* __restrict__ knots,
                             const float* __restrict__ ca,
                             const float* __restrict__ cb,
                             const float* __restrict__ cc,
                             const float* __restrict__ cd,
                             const float* __restrict__ W1,
                             const float* __restrict__ b1,
                             const float* __restrict__ W2,
                             const float* __restrict__ b2,
                             const float* __restrict__ W3,
                             const float* __restrict__ b3,
                             const float* __restrict__ W4,
                             const float* __restrict__ b4,
                             const float* __restrict__ w1f,
                             const float* __restrict__ w3f,
                             float* __restrict__ out) {
    __shared__ float lds_part[NWAVES * 256];           // per-wave 16x16 f32 partials (8 KB)
    __shared__ float lds_h[ROWS_PER_BLOCK * 16];       // h, N padded to 16 (1 KB)
    __shared__ float lds_out[ROWS_PER_BLOCK * OUTSTR]; // spline outputs [16][512] padded (33 KB)

    const int tid   = threadIdx.x;
    const int wave  = tid >> 5;
    const int lane  = tid & 31;
    const int mrow  = lane & 15;            // A-matrix M row / B-matrix N column for this lane
    const int khalf = (lane >> 4) ? 2 : 0;  // lanes 16-31 hold K = {2,3} of each 4-slice

    const int rowg = blockIdx.x * ROWS_PER_BLOCK + mrow;
    const size_t xrow = (size_t)rowg * FDIM;

    // Branch-free B fallback (direct path): clamped column + 0/1 mask.
    const float bmask = (mrow < HDIM) ? 1.0f : 0.0f;
    const int   ncol  = (mrow < HDIM) ? mrow : 0;

    auto loadA1 = [&](int s) -> v2f {  // A fragment: x[rowg][4s + khalf .. +1], 8B aligned
        return *(const v2f*)(x + xrow + s * 4 + khalf);
    };
    auto loadB1 = [&](int s) -> v2f {  // B fragment: W1 slice, frag stream or direct
        v2f b;
        if (USE_FRAG) {
            b = *(const v2f*)(w1f + s * 64 + lane * 2);
        } else {
            const int k0 = s * 4;
            b[0] = W1[(k0 + khalf) * HDIM + ncol] * bmask;
            b[1] = W1[(k0 + khalf + 1) * HDIM + ncol] * bmask;
        }
        return b;
    };

    // ---------------- Phase 1 ----------------
    // Wave w handles steps s = w + 8*j, j = 0..130 uniform (131 steps); waves 0,1
    // additionally do j = 131 (s = 1048+wave). Flat 4-step body: all loads are
    // body-local temporaries (no loop-carried fragment registers), so WMMAs
    // accumulate in place and the scheduler clusters loads with staggered waits.
    {
        v8f acc0 = {0.f, 0.f, 0.f, 0.f, 0.f, 0.f, 0.f, 0.f};
        v8f acc1 = {0.f, 0.f, 0.f, 0.f, 0.f, 0.f, 0.f, 0.f};
#pragma unroll 1
        for (int i = 0; i < 32; ++i) {
            const int s0 = wave + NWAVES * (4 * i);
            const int s1 = s0 + NWAVES;
            const int s2 = s0 + 2 * NWAVES;
            const int s3 = s0 + 3 * NWAVES;
            v2f A0 = loadA1(s0), B0 = loadB1(s0);
            v2f A1 = loadA1(s1), B1 = loadB1(s1);
            v2f A2 = loadA1(s2), B2 = loadB1(s2);
            v2f A3 = loadA1(s3), B3 = loadB1(s3);
            acc0 = WMMA_F32_4(A0, B0, acc0);
            acc1 = WMMA_F32_4(A1, B1, acc1);
            acc0 = WMMA_F32_4(A2, B2, acc0);
            acc1 = WMMA_F32_4(A3, B3, acc1);
        }
        {   // drain: j = 128, 129, 130 for all waves; j = 131 for waves 0,1
            v2f A0 = loadA1(wave + NWAVES * 128), B0 = loadB1(wave + NWAVES * 128);
            v2f A1 = loadA1(wave + NWAVES * 129), B1 = loadB1(wave + NWAVES * 129);
            v2f A2 = loadA1(wave + NWAVES * 130), B2 = loadB1(wave + NWAVES * 130);
            acc0 = WMMA_F32_4(A0, B0, acc0);
            acc1 = WMMA_F32_4(A1, B1, acc1);
            acc0 = WMMA_F32_4(A2, B2, acc0);
            if (wave < 2) {  // wave-uniform: EXEC all-ones (or branch skipped entirely)
                v2f A3 = loadA1(1048 + wave), B3 = loadB1(1048 + wave);
                acc1 = WMMA_F32_4(A3, B3, acc1);
            }
        }
#pragma unroll
        for (int r = 0; r < 8; ++r)
            lds_part[wave * 256 + lane * 8 + r] = acc0[r] + acc1[r];
    }
    __syncthreads();

    // Cross-wave reduction of the 16x16 accumulator; bias + leaky ReLU.
    {
        const int m = tid >> 4, n = tid & 15;
        const int lsel = n + 16 * (m >> 3);  // C layout: VGPR r = rows r (lanes 0-15) / r+8 (16-31)
        const int r = m & 7;
        float s = 0.f;
#pragma unroll
        for (int w = 0; w < NWAVES; ++w) s += lds_part[w * 256 + lsel * 8 + r];
        float hv = 0.f;
        if (n < HDIM) hv = leaky01(s + b1[n]);
        lds_h[m * 16 + n] = hv;
    }
    __syncthreads();

    // ---------------- Phase 2 ----------------
    {
        const float scale = (float)(FDIM - 1);
        for (int i = 0; i < 32; ++i) {
            const int item = i * 256 + tid;
            const int m = item >> 9;
            const int t = item & (TDIM - 1);
            float z = b2[t];
#pragma unroll
            for (int j = 0; j < HDIM; ++j) z += lds_h[m * 16 + j] * W2[j * TDIM + t];
            const float qv = 1.0f / (1.0f + __expf(-z));
            // searchsorted(knots, q, 'right') - 1, nudged against the actual knot values
            int i0 = (int)floorf(qv * scale);
            i0 = i0 < 0 ? 0 : (i0 > FDIM - 2 ? FDIM - 2 : i0);
            while (i0 < FDIM - 2 && knots[i0 + 1] <= qv) ++i0;
            while (i0 > 0 && knots[i0] > qv) --i0;
            const float dx = qv - knots[i0];
            const size_t g = (size_t)(blockIdx.x * ROWS_PER_BLOCK + m) * (FDIM - 1) + i0;
            const float av = ca[g], bv = cb[g], cv = cc[g], dv = cd[g];
            lds_out[m * OUTSTR + t] = av + dx * (bv + dx * (cv + dx * dv));
        }
    }
    __syncthreads();

    // ---------------- Phase 3 ----------------
    // Exactly 16 steps per wave (128/8): fixed 8x2 loop, no guards.
    {
        auto loadB3 = [&](int s) -> v2f {
            v2f b;
            if (USE_FRAG) {
                b = *(const v2f*)(w3f + s * 64 + lane * 2);
            } else {
                const int k0 = s * 4;
                b[0] = W3[(k0 + khalf) * HDIM + ncol] * bmask;
                b[1] = W3[(k0 + khalf + 1) * HDIM + ncol] * bmask;
            }
            return b;
        };
        v8f acc0 = {0.f, 0.f, 0.f, 0.f, 0.f, 0.f, 0.f, 0.f};
        v8f acc1 = {0.f, 0.f, 0.f, 0.f, 0.f, 0.f, 0.f, 0.f};
#pragma unroll 2
        for (int i = 0; i < 8; ++i) {
            const int s0 = wave + 16 * i;
            const int s1 = s0 + NWAVES;
            v2f a = *(const v2f*)(&lds_out[mrow * OUTSTR + s0 * 4 + khalf]);
            v2f b = loadB3(s0);
            acc0 = WMMA_F32_4(a, b, acc0);
            a = *(const v2f*)(&lds_out[mrow * OUTSTR + s1 * 4 + khalf]);
            b = loadB3(s1);
            acc1 = WMMA_F32_4(a, b, acc1);
        }
#pragma unroll
        for (int r = 0; r < 8; ++r)
            lds_part[wave * 256 + lane * 8 + r] = acc0[r] + acc1[r];
    }
    __syncthreads();

    {
        const int m = tid >> 4, n = tid & 15;
        const int lsel = n + 16 * (m >> 3);
        const int r = m & 7;
        float s = 0.f;
#pragma unroll
        for (int w = 0; w < NWAVES; ++w) s += lds_part[w * 256 + lsel * 8 + r];
        float h2 = 0.f;
        if (n < HDIM) h2 = leaky01(s + b3[n]);
        lds_h[m * 16 + n] = h2;
    }
    __syncthreads();

    if (tid < ROWS_PER_BLOCK) {
        float acc = b4[0];
#pragma unroll
        for (int j = 0; j < HDIM; ++j) acc += lds_h[tid * 16 + j] * W4[j];
        out[blockIdx.x * ROWS_PER_BLOCK + tid] = acc;
    }
}

extern "C" void kernel_launch(void* const* d_in, const int* in_sizes, int n_in,
                              void* d_out, int out_size, void* d_ws, size_t ws_size,
                              hipStream_t stream) {
    (void)in_sizes; (void)n_in; (void)out_size;
    const float* x     = (const float*)d_in[0];
    const float* knots = (const float*)d_in[1];
    const float* ca    = (const float*)d_in[2];
    const float* cb    = (const float*)d_in[3];
    const float* cc    = (const float*)d_in[4];
    const float* cd    = (const float*)d_in[5];
    const float* W1    = (const float*)d_in[6];
    const float* b1    = (const float*)d_in[7];
    const float* W2    = (const float*)d_in[8];
    const float* b2    = (const float*)d_in[9];
    const float* W3    = (const float*)d_in[10];
    const float* b3    = (const float*)d_in[11];
    const float* W4    = (const float*)d_in[12];
    const float* b4    = (const float*)d_in[13];
    float* out = (float*)d_out;

    const dim3 grid(BATCH / ROWS_PER_BLOCK);  // 512 workgroups x 16 batch rows
    const dim3 block(BLOCK_THREADS);          // 8 waves (wave32)

    const size_t frag_bytes = (size_t)(NSTEP1 * 64 + NSTEP3 * 64) * sizeof(float);
    if (ws_size >= frag_bytes) {
        float* w1f = (float*)d_ws;
        float* w3f = w1f + NSTEP1 * 64;
        const int prep_items = NSTEP1 * 64;  // >= NSTEP3 * 64
        ANN_prep_frags<<<(prep_items + 255) / 256, 256, 0, stream>>>(W1, W3, w1f, w3f);
        ANN_spline_fused_kernel<true><<<grid, block, 0, stream>>>(
            x, knots, ca, cb, cc, cd, W1, b1, W2, b2, W3, b3, W4, b4, w1f, w3f, out);
    } else {
        ANN_spline_fused_kernel<false><<<grid, block, 0, stream>>>(
            x, knots, ca, cb, cc, cd, W1, b1, W2, b2, W3, b3, W4, b4, nullptr, nullptr, out);
    }
}